// SwinMVS_64372969832931
// MI455X (gfx1250) — compile-verified
//
#include <hip/hip_runtime.h>

typedef __attribute__((ext_vector_type(2))) float v2f;
typedef __attribute__((ext_vector_type(8))) float v8f;

#define CC   32
#define HH   96
#define WW   96
#define DD   96
#define HWSZ (HH * WW)          // 9216
#define CHW  (CC * HWSZ)        // 294912
#define DHW  (DD * HWSZ)        // 884736

// ---------------------------------------------------------------------------
// Kernel 1: compute proj_v = src_projs[v] @ inv(ref_proj); store rot rows +
// trans per view into ws_proj[v*12 + i*4 + j] (i<3, j<4, row-major).
// ---------------------------------------------------------------------------
__global__ void mvs_prep_proj(const float* __restrict__ ref_proj,
                              const float* __restrict__ src_projs,
                              float* __restrict__ ws_proj) {
    if (threadIdx.x != 0 || blockIdx.x != 0) return;
    float m[16], inv[16];
    #pragma unroll
    for (int i = 0; i < 16; ++i) m[i] = ref_proj[i];

    // Cofactor-based 4x4 inverse (row-major consistent).
    inv[0]  =  m[5]*m[10]*m[15] - m[5]*m[11]*m[14] - m[9]*m[6]*m[15] + m[9]*m[7]*m[14] + m[13]*m[6]*m[11] - m[13]*m[7]*m[10];
    inv[4]  = -m[4]*m[10]*m[15] + m[4]*m[11]*m[14] + m[8]*m[6]*m[15] - m[8]*m[7]*m[14] - m[12]*m[6]*m[11] + m[12]*m[7]*m[10];
    inv[8]  =  m[4]*m[9]*m[15]  - m[4]*m[11]*m[13] - m[8]*m[5]*m[15] + m[8]*m[7]*m[13] + m[12]*m[5]*m[11] - m[12]*m[7]*m[9];
    inv[12] = -m[4]*m[9]*m[14]  + m[4]*m[10]*m[13] + m[8]*m[5]*m[14] - m[8]*m[6]*m[13] - m[12]*m[5]*m[10] + m[12]*m[6]*m[9];
    inv[1]  = -m[1]*m[10]*m[15] + m[1]*m[11]*m[14] + m[9]*m[2]*m[15] - m[9]*m[3]*m[14] - m[13]*m[2]*m[11] + m[13]*m[3]*m[10];
    inv[5]  =  m[0]*m[10]*m[15] - m[0]*m[11]*m[14] - m[8]*m[2]*m[15] + m[8]*m[3]*m[14] + m[12]*m[2]*m[11] - m[12]*m[3]*m[10];
    inv[9]  = -m[0]*m[9]*m[15]  + m[0]*m[11]*m[13] + m[8]*m[1]*m[15] - m[8]*m[3]*m[13] - m[12]*m[1]*m[11] + m[12]*m[3]*m[9];
    inv[13] =  m[0]*m[9]*m[14]  - m[0]*m[10]*m[13] - m[8]*m[1]*m[14] + m[8]*m[2]*m[13] + m[12]*m[1]*m[10] - m[12]*m[2]*m[9];
    inv[2]  =  m[1]*m[6]*m[15]  - m[1]*m[7]*m[14]  - m[5]*m[2]*m[15] + m[5]*m[3]*m[14] + m[13]*m[2]*m[7]  - m[13]*m[3]*m[6];
    inv[6]  = -m[0]*m[6]*m[15]  + m[0]*m[7]*m[14]  + m[4]*m[2]*m[15] - m[4]*m[3]*m[14] - m[12]*m[2]*m[7]  + m[12]*m[3]*m[6];
    inv[10] =  m[0]*m[5]*m[15]  - m[0]*m[7]*m[13]  - m[4]*m[1]*m[15] + m[4]*m[3]*m[13] + m[12]*m[1]*m[7]  - m[12]*m[3]*m[5];
    inv[14] = -m[0]*m[5]*m[14]  + m[0]*m[6]*m[13]  + m[4]*m[1]*m[14] - m[4]*m[2]*m[13] - m[12]*m[1]*m[6]  + m[12]*m[2]*m[5];
    inv[3]  = -m[1]*m[6]*m[11]  + m[1]*m[7]*m[10]  + m[5]*m[2]*m[11] - m[5]*m[3]*m[10] - m[9]*m[2]*m[7]   + m[9]*m[3]*m[6];
    inv[7]  =  m[0]*m[6]*m[11]  - m[0]*m[7]*m[10]  - m[4]*m[2]*m[11] + m[4]*m[3]*m[10] + m[8]*m[2]*m[7]   - m[8]*m[3]*m[6];
    inv[11] = -m[0]*m[5]*m[11]  + m[0]*m[7]*m[9]   + m[4]*m[1]*m[11] - m[4]*m[3]*m[9]  - m[8]*m[1]*m[7]   + m[8]*m[3]*m[5];
    inv[15] =  m[0]*m[5]*m[10]  - m[0]*m[6]*m[9]   - m[4]*m[1]*m[10] + m[4]*m[2]*m[9]  + m[8]*m[1]*m[6]   - m[8]*m[2]*m[5];
    float det = m[0]*inv[0] + m[1]*inv[4] + m[2]*inv[8] + m[3]*inv[12];
    float rdet = 1.0f / det;
    #pragma unroll
    for (int i = 0; i < 16; ++i) inv[i] *= rdet;

    for (int v = 0; v < 2; ++v) {
        const float* S = src_projs + v * 16;
        for (int i = 0; i < 3; ++i)
            for (int j = 0; j < 4; ++j) {
                float acc = 0.0f;
                for (int k = 0; k < 4; ++k) acc = fmaf(S[i*4 + k], inv[k*4 + j], acc);
                ws_proj[v*12 + i*4 + j] = acc;
            }
    }
}

// ---------------------------------------------------------------------------
// Kernel 2: base[n][p] = sum_k rot_v[i][k] * xyz1[k][p], n = 3v+i, via
// V_WMMA_F32_16X16X4_F32.  One wave per 16-pixel tile.
//   A (16x4, MxK): lanes 0-15 hold (K0,K1)=(x,y); lanes 16-31 hold (K2,K3)=(1,0)
//   B (4x16, KxN): VGPR0 = row K(0 + 2*hi), VGPR1 = row K(1 + 2*hi)
//   D (16x16):     VGPR r, lane l: M = r + 8*(l>>4), N = l & 15
// ---------------------------------------------------------------------------
__global__ __launch_bounds__(32) void mvs_prep_base(const float* __restrict__ ws_proj,
                                                    float* __restrict__ ws_base) {
    const int lane = threadIdx.x;        // 0..31, one full wave32
    const int m    = lane & 15;
    const int hi   = lane >> 4;
    const int pixbase = blockIdx.x * 16;
    const int pix  = pixbase + m;
    const float fx = (float)(pix % WW);
    const float fy = (float)(pix / WW);

    v2f a;
    a.x = hi ? 1.0f : fx;                // K2 = 1        | K0 = x
    a.y = hi ? 0.0f : fy;                // K3 = 0        | K1 = y

    // B column n = 3v+i holds rot_v[i][k]; columns 6..15 are zero.
    const int  n   = m;
    const int  v   = (n >= 3) ? 1 : 0;
    const int  i   = n - 3 * v;
    const bool use = (n < 6);
    const float* P = ws_proj + (use ? (v * 12 + i * 4) : 0);
    float bx = use ? (hi ? P[2] : P[0]) : 0.0f;   // K2 | K0
    float by = use ? (hi ? 0.0f : P[1]) : 0.0f;   // K3 | K1
    v2f b; b.x = bx; b.y = by;

    v8f cz = {};
    v8f d = __builtin_amdgcn_wmma_f32_16x16x4_f32(false, a, false, b,
                                                  (short)0, cz, false, false);

    if (use) {
        const int pb = pixbase + 8 * hi;          // M = r + 8*hi
        #pragma unroll
        for (int r = 0; r < 8; ++r)
            ws_base[n * HWSZ + pb + r] = d[r];
    }
}

// ---------------------------------------------------------------------------
// Kernel 3: streaming cost-volume build. One thread per (d, h, w).
// idx = d*HW + hw == inner output index, so stores are perfectly coalesced.
// ---------------------------------------------------------------------------
__global__ __launch_bounds__(256) void mvs_cost_main(const float* __restrict__ ref,
                                                     const float* __restrict__ src,
                                                     const float* __restrict__ depth_values,
                                                     const float* __restrict__ ws_proj,
                                                     const float* __restrict__ ws_base,
                                                     float* __restrict__ out) {
    const int idx = blockIdx.x * 256 + threadIdx.x;     // = d*HW + hw
    if (idx >= DHW) return;
    const int d  = idx / HWSZ;
    const int hw = idx - d * HWSZ;
    const float depth = depth_values[d];                // wave-uniform

    float wgt[2][4];
    int   off[2][4];

    #pragma unroll
    for (int v = 0; v < 2; ++v) {
        const float b0 = ws_base[(3*v + 0) * HWSZ + hw];
        const float b1 = ws_base[(3*v + 1) * HWSZ + hw];
        const float b2 = ws_base[(3*v + 2) * HWSZ + hw];
        const float t0 = ws_proj[v*12 + 3];
        const float t1 = ws_proj[v*12 + 7];
        const float t2 = ws_proj[v*12 + 11];

        const float z  = fmaf(t2, depth, b2) + 1e-10f;
        const float rz = 1.0f / z;
        const float px = fmaf(t0, depth, b0) * rz;
        const float py = fmaf(t1, depth, b1) * rz;

        const float x0f = floorf(px), y0f = floorf(py);
        const float wx = px - x0f, wy = py - y0f;
        const int x0 = (int)x0f, y0 = (int)y0f;
        const int x1 = x0 + 1,   y1 = y0 + 1;

        const float okx0 = (x0 >= 0 && x0 < WW) ? 1.0f : 0.0f;
        const float okx1 = (x1 >= 0 && x1 < WW) ? 1.0f : 0.0f;
        const float oky0 = (y0 >= 0 && y0 < HH) ? 1.0f : 0.0f;
        const float oky1 = (y1 >= 0 && y1 < HH) ? 1.0f : 0.0f;
        const int x0c = min(max(x0, 0), WW - 1);
        const int x1c = min(max(x1, 0), WW - 1);
        const int y0c = min(max(y0, 0), HH - 1);
        const int y1c = min(max(y1, 0), HH - 1);

        const float fx0 = 1.0f - wx, fx1 = wx;
        const float fy0 = 1.0f - wy, fy1 = wy;
        wgt[v][0] = fx0 * fy0 * okx0 * oky0;  off[v][0] = y0c * WW + x0c;
        wgt[v][1] = fx1 * fy0 * okx1 * oky0;  off[v][1] = y0c * WW + x1c;
        wgt[v][2] = fx0 * fy1 * okx0 * oky1;  off[v][2] = y1c * WW + x0c;
        wgt[v][3] = fx1 * fy1 * okx1 * oky1;  off[v][3] = y1c * WW + x1c;
    }

    const float* __restrict__ s0 = src;           // view 0 features [C,H,W]
    const float* __restrict__ s1 = src + CHW;     // view 1 features
    const float inv3 = 1.0f / 3.0f;

    #pragma unroll 4
    for (int c = 0; c < CC; ++c) {
        const int cb = c * HWSZ;
        // hint the next channel plane of ref into cache (global_prefetch_b8)
        if (c + 1 < CC) __builtin_prefetch(ref + cb + HWSZ + hw, 0, 1);

        const float r = ref[cb + hw];

        float w0 = wgt[0][0] * s0[cb + off[0][0]];
        w0 = fmaf(wgt[0][1], s0[cb + off[0][1]], w0);
        w0 = fmaf(wgt[0][2], s0[cb + off[0][2]], w0);
        w0 = fmaf(wgt[0][3], s0[cb + off[0][3]], w0);

        float w1 = wgt[1][0] * s1[cb + off[1][0]];
        w1 = fmaf(wgt[1][1], s1[cb + off[1][1]], w1);
        w1 = fmaf(wgt[1][2], s1[cb + off[1][2]], w1);
        w1 = fmaf(wgt[1][3], s1[cb + off[1][3]], w1);

        out[c * DHW + idx]            = r * (2.0f * inv3);
        out[(CC + c) * DHW + idx]     = (w0 + w1) * inv3;
        out[(2*CC + c) * DHW + idx]   = (fabsf(r - w0) + fabsf(r - w1)) * inv3;
    }
}

// ---------------------------------------------------------------------------
extern "C" void kernel_launch(void* const* d_in, const int* in_sizes, int n_in,
                              void* d_out, int out_size, void* d_ws, size_t ws_size,
                              hipStream_t stream) {
    const float* ref       = (const float*)d_in[0];   // [1,32,96,96]
    const float* src       = (const float*)d_in[1];   // [2,1,32,96,96]
    const float* ref_proj  = (const float*)d_in[2];   // [1,4,4]
    const float* src_projs = (const float*)d_in[3];   // [2,1,4,4]
    const float* depth     = (const float*)d_in[4];   // [1,96]
    float* out = (float*)d_out;

    float* ws_proj = (float*)d_ws;        // 24 floats (padded to 32)
    float* ws_base = ws_proj + 32;        // 6 * 9216 floats

    mvs_prep_proj<<<1, 32, 0, stream>>>(ref_proj, src_projs, ws_proj);
    mvs_prep_base<<<HWSZ / 16, 32, 0, stream>>>(ws_proj, ws_base);
    mvs_cost_main<<<DHW / 256, 256, 0, stream>>>(ref, src, depth, ws_proj, ws_base, out);
}